// e_prod_KernelChannels5_3728031613246
// MI455X (gfx1250) — compile-verified
//
#include <hip/hip_runtime.h>
#include <stdint.h>

// Problem constants (B=2, C=8 -> 16 planes of H x W)
#define H_      256
#define W_      512
#define NPLANE  16
#define TH      16
#define TW      64
#define SH      (TH + 4)       // 20 rows  (halo 2)
#define SW      (TW + 4)       // 68 cols  (halo 2)
#define TILES_X (W_ / TW)      // 8
#define TILES_Y (H_ / TH)      // 16

typedef __attribute__((ext_vector_type(2))) float v2f;

// ---- CDNA5 async global -> LDS copy (ASYNCcnt path) --------------------
// VDST = per-lane LDS byte address (low 32 bits of generic pointer == LDS
// offset per the gfx1250 aperture rule), VADDR = 64-bit global address.
__device__ __forceinline__ void async_ld_b32(uint32_t lds_addr, const float* g) {
    asm volatile("global_load_async_to_lds_b32 %0, %1, off"
                 :: "v"(lds_addr), "v"(g) : "memory");
}
__device__ __forceinline__ void wait_asynccnt0() {
    asm volatile("s_wait_asynccnt 0" ::: "memory");
}

__device__ __forceinline__ float inb(int y, int x) {
    return ((unsigned)y < (unsigned)H_ && (unsigned)x < (unsigned)W_) ? 1.0f : 0.0f;
}

// argmax with first-index tie-break: later candidate wins only on strict >
__device__ __forceinline__ float sel2(float a0, float c0, float a1, float c1) {
    return (c1 > c0) ? a1 : a0;
}
__device__ __forceinline__ float sel3(float a0, float c0, float a1, float c1,
                                      float a2, float c2) {
    float a = a0, c = c0;
    if (c1 > c) { c = c1; a = a1; }
    if (c2 > c) { a = a2; }
    return a;
}

// Compute the 25 output channels for one pixel.
// Se/Sc point at the pixel's center inside the LDS tiles.
// BORDER=false: all radius-1 gates are provably 1 -> no indicator math.
template <bool BORDER>
__device__ __forceinline__ void compute25(const float* __restrict__ Se,
                                          const float* __restrict__ Sc,
                                          int gi, int gj, float v[25]) {
    float iNm = 1.f, iN = 1.f, iNp = 1.f, iWm = 1.f, iE = 1.f,
          iSm = 1.f, iS = 1.f, iSp = 1.f;
    if (BORDER) {
        iNm = inb(gi - 1, gj - 1); iN = inb(gi - 1, gj); iNp = inb(gi - 1, gj + 1);
        iWm = inb(gi,     gj - 1);                        iE  = inb(gi,     gj + 1);
        iSm = inb(gi + 1, gj - 1); iS = inb(gi + 1, gj); iSp = inb(gi + 1, gj + 1);
    }
#define EV(dy, dx) Se[(dy) * SW + (dx)]
#define CV(dy, dx) Sc[(dy) * SW + (dx)]
#define GT(x, g)   (BORDER ? ((x) * (g)) : (x))

    // e: inner ring (E(0..8) ex 4)
    const float e00v = EV(-1,-1), e01v = EV(-1, 0), e02v = EV(-1, 1);
    const float e10v = EV( 0,-1),                   e12v = EV( 0, 1);
    const float e20v = EV( 1,-1), e21v = EV( 1, 0), e22v = EV( 1, 1);
    // e: outer ring samples actually used
    const float eA = EV(-2,-2), eB = EV(-2,-1), eC = EV(-2, 0), eD = EV(-2, 1), eF = EV(-2, 2);
    const float eG = EV(-1,-2), eHh = EV(0,-2), eI = EV( 1,-2);
    const float eJ = EV(-1, 2), eK  = EV(0, 2), eL = EV( 1, 2);
    const float eM = EV( 2,-2), eN = EV( 2,-1), eO = EV( 2, 0), eP = EV( 2, 1), eQ = EV( 2, 2);
    // ce: inner ring + center (center needed by the e24 Cp(3,1,2) anomaly)
    const float c00v = CV(-1,-1), c01v = CV(-1, 0), c02v = CV(-1, 1);
    const float c10v = CV( 0,-1), cZZ  = CV( 0, 0), c12v = CV( 0, 1);
    const float c20v = CV( 1,-1), c21v = CV( 1, 0), c22v = CV( 1, 1);
    // ce: outer ring samples actually used (corners unused)
    const float cB = CV(-2,-1), cC = CV(-2, 0), cD = CV(-2, 1);
    const float cG = CV(-1,-2), cH = CV(0,-2), cI = CV( 1,-2);
    const float cJ = CV(-1, 2), cK = CV(0, 2), cL = CV( 1, 2);
    const float cN = CV( 2,-1), cO = CV( 2, 0), cP = CV( 2, 1);

    // corners
    v[0]  = e00v * GT(eA, iNm);                                     // e00
    v[4]  = e02v * GT(eF, iNp);                                     // e04
    v[20] = e20v * GT(eM, iSm);                                     // e40
    v[24] = e22v * GT(eQ, iSp);                                     // e44
    // top row
    v[1] = sel2(GT(e00v*eB, iNm), GT(c00v*cB, iNm),
                GT(e01v*eB, iN ), GT(c01v*cB, iN ));                // e01
    v[2] = sel3(GT(e00v*eC, iNm), GT(c00v*cC, iNm),
                GT(e01v*eC, iN ), GT(c01v*cC, iN ),
                GT(e02v*eC, iNp), GT(c02v*cC, iNp));                // e02
    v[3] = sel2(GT(e01v*eD, iN ), GT(c01v*cD, iN ),
                GT(e02v*eD, iNp), GT(c02v*cD, iNp));                // e03
    // left column
    v[5]  = sel2(GT(e00v*eG, iNm),  GT(c00v*cG, iNm),
                 GT(e10v*eG, iWm),  GT(c10v*cG, iWm));              // e10
    v[10] = sel3(GT(e00v*eHh, iNm), GT(c00v*cH, iNm),
                 GT(e10v*eHh, iWm), GT(c10v*cH, iWm),
                 GT(e20v*eHh, iSm), GT(c20v*cH, iSm));              // e20
    v[15] = sel2(GT(e10v*eI, iWm),  GT(c10v*cI, iWm),
                 GT(e20v*eI, iSm),  GT(c20v*cI, iSm));              // e30
    // right column
    v[9]  = sel2(GT(e02v*eJ, iNp), GT(c02v*cJ, iNp),
                 GT(e12v*eJ, iE ), GT(c12v*cJ, iE ));               // e14
    v[14] = sel3(GT(e02v*eK, iNp), GT(c02v*cK, iNp),
                 GT(e12v*eK, iE ), GT(c12v*cZZ, iE ),  /* ref anomaly: Cp(3,1,2) */
                 GT(e22v*eK, iSp), GT(c22v*cK, iSp));               // e24
    v[19] = sel2(GT(e12v*eL, iE ), GT(c12v*cL, iE ),
                 GT(e22v*eL, iSp), GT(c22v*cL, iSp));               // e34
    // bottom row
    v[21] = sel2(GT(e20v*eN, iSm), GT(c20v*cN, iSm),
                 GT(e21v*eN, iS ), GT(c21v*cN, iS ));               // e41
    v[22] = sel3(GT(e20v*eO, iSm), GT(c20v*cO, iSm),
                 GT(e21v*eO, iS ), GT(c21v*cO, iS ),
                 GT(e22v*eO, iSp), GT(c22v*cO, iSp));               // e42
    v[23] = sel2(GT(e21v*eP, iS ), GT(c21v*cP, iS ),
                 GT(e22v*eP, iSp), GT(c22v*cP, iSp));               // e43
    // direct E(k) channels (middle of 5x5)
    v[6]  = e00v; v[7]  = e01v; v[8]  = e02v;
    v[11] = e10v; v[12] = 1.0f; v[13] = e12v;
    v[16] = e20v; v[17] = e21v; v[18] = e22v;

#undef EV
#undef CV
#undef GT
}

template <bool BORDER>
__global__ __launch_bounds__(256)
void eprod25_kernel(const float* __restrict__ e,
                    const float* __restrict__ ce,
                    float* __restrict__ out) {
    __shared__ float se[SH * SW];
    __shared__ float sc[SH * SW];

    const int tid = threadIdx.x;

    // tile decode
    int tix, tiy, plane;
    if (BORDER) {
        // 44 border tiles per plane: tiy in {0,15} or tix in {0,7}
        int blk  = blockIdx.x;
        int bidx = blk % 44;
        plane    = blk / 44;
        if (bidx < 8)       { tiy = 0;  tix = bidx; }
        else if (bidx < 16) { tiy = 15; tix = bidx - 8; }
        else {
            int r = bidx - 16;            // 0..27
            tix = (r < 14) ? 0 : 7;
            tiy = 1 + (r % 14);
        }
    } else {
        // 84 interior tiles per plane: tix in 1..6, tiy in 1..14
        int blk = blockIdx.x;
        tix = 1 + blk % 6;  blk /= 6;
        tiy = 1 + blk % 14; blk /= 14;
        plane = blk;
    }
    const int i0 = tiy * TH;
    const int j0 = tix * TW;

    const size_t HW = (size_t)H_ * W_;
    const float* ep = e  + (size_t)plane * HW;
    const float* cp = ce + (size_t)plane * HW;
    float*       op = out + (size_t)plane * 25 * HW;

    if (BORDER) {
        // Zero-fill provides the inner zero-padding level at image borders.
        for (int l = tid; l < SH * SW; l += 256) { se[l] = 0.0f; sc[l] = 0.0f; }
        __syncthreads();
        for (int l = tid; l < SH * SW; l += 256) {
            const int r  = l / SW;
            const int c  = l - r * SW;
            const int gi = i0 - 2 + r;
            const int gj = j0 - 2 + c;
            if ((unsigned)gi < (unsigned)H_ && (unsigned)gj < (unsigned)W_) {
                const size_t off = (size_t)gi * W_ + gj;
                async_ld_b32((uint32_t)(uintptr_t)(&se[l]), ep + off);
                async_ld_b32((uint32_t)(uintptr_t)(&sc[l]), cp + off);
            }
        }
    } else {
        // Interior tile: halo is fully in-bounds -> straight async gather,
        // no zero-fill, no bounds branch, one barrier total.
        for (int l = tid; l < SH * SW; l += 256) {
            const int r   = l / SW;
            const int c   = l - r * SW;
            const size_t off = (size_t)(i0 - 2 + r) * W_ + (j0 - 2 + c);
            async_ld_b32((uint32_t)(uintptr_t)(&se[l]), ep + off);
            async_ld_b32((uint32_t)(uintptr_t)(&sc[l]), cp + off);
        }
    }
    wait_asynccnt0();
    __syncthreads();

    // Thread -> 2 consecutive columns x 2 rows; a wave32 covers 64
    // consecutive columns, so each channel store is 256B contiguous (b64/lane).
    const int lxp = (tid & 31) * 2;   // local col of first pixel (even)
    const int ty  = tid >> 5;         // 0..7

    #pragma unroll
    for (int ry = 0; ry < 2; ++ry) {
        const int ly = ty * 2 + ry;
        const int gi = i0 + ly;
        const int gj = j0 + lxp;
        const float* Se = &se[(ly + 2) * SW + (lxp + 2)];
        const float* Sc = &sc[(ly + 2) * SW + (lxp + 2)];

        float va[25], vb[25];
        compute25<BORDER>(Se,     Sc,     gi, gj,     va);
        compute25<BORDER>(Se + 1, Sc + 1, gi, gj + 1, vb);

        // Streaming output (200 MB, never re-read): non-temporal b64 stores
        // keep the 16 MB of inputs resident in the 192 MB L2.
        float* o = op + (size_t)gi * W_ + gj;   // 8B-aligned (gj even)
        #pragma unroll
        for (int ch = 0; ch < 25; ++ch) {
            v2f pv; pv.x = va[ch]; pv.y = vb[ch];
            __builtin_nontemporal_store(pv, (v2f*)(o + (size_t)ch * HW));
        }
    }
}

extern "C" void kernel_launch(void* const* d_in, const int* in_sizes, int n_in,
                              void* d_out, int out_size, void* d_ws, size_t ws_size,
                              hipStream_t stream) {
    (void)in_sizes; (void)n_in; (void)out_size; (void)d_ws; (void)ws_size;
    const float* e  = (const float*)d_in[0];
    const float* ce = (const float*)d_in[1];
    float* out = (float*)d_out;

    // 84 interior + 44 border tiles per plane (16 planes), 8 wave32 per block
    eprod25_kernel<false><<<dim3(NPLANE * 14 * 6), dim3(256), 0, stream>>>(e, ce, out);
    eprod25_kernel<true ><<<dim3(NPLANE * 44),     dim3(256), 0, stream>>>(e, ce, out);
}